// MessagePassing_12558484374174
// MI455X (gfx1250) — compile-verified
//
#include <hip/hip_runtime.h>

#define GNN_FEAT      128          // floats per node row
#define GNN_F4        32           // float4 per node row (128/4)

// ---------------------------------------------------------------------------
// Utility: zero int / float buffers
// ---------------------------------------------------------------------------
__global__ void gnn_zero_i32(int* __restrict__ p, int n) {
    int i = blockIdx.x * blockDim.x + threadIdx.x;
    if (i < n) p[i] = 0;
}

__global__ void gnn_zero_f32(float* __restrict__ p, int n) {
    int i = blockIdx.x * blockDim.x + threadIdx.x;
    if (i < n) p[i] = 0.0f;
}

// ---------------------------------------------------------------------------
// Phase 1: histogram of destination node ids (counts accumulated into start[])
// ---------------------------------------------------------------------------
__global__ void gnn_hist(const int* __restrict__ edges,   // [nE][2] (dst, src)
                         int* __restrict__ counts, int nE) {
    int e = blockIdx.x * blockDim.x + threadIdx.x;
    if (e < nE) {
        const int2 de = ((const int2*)edges)[e];          // global_load_b64
        atomicAdd(&counts[de.x], 1);
    }
}

// ---------------------------------------------------------------------------
// Phase 2: single-block exclusive scan over counts (in-place -> start[]),
// also writes cursor[] copy and start[n] = total.  n = 10000 -> 10 chunks.
// ---------------------------------------------------------------------------
__global__ __launch_bounds__(1024)
void gnn_scan(int* __restrict__ start, int* __restrict__ cursor, int n) {
    __shared__ int buf[1024];
    __shared__ int carry_s;
    const int tid = threadIdx.x;
    if (tid == 0) carry_s = 0;
    __syncthreads();

    for (int chunk = 0; chunk < n; chunk += 1024) {
        const int idx = chunk + tid;
        const int val = (idx < n) ? start[idx] : 0;
        buf[tid] = val;
        __syncthreads();
        // Hillis-Steele inclusive scan in LDS
        for (int off = 1; off < 1024; off <<= 1) {
            int t = (tid >= off) ? buf[tid - off] : 0;
            __syncthreads();
            buf[tid] += t;
            __syncthreads();
        }
        const int incl  = buf[tid];
        const int excl  = incl - val + carry_s;
        if (idx < n) { start[idx] = excl; cursor[idx] = excl; }
        const int total = buf[1023] + carry_s;
        __syncthreads();
        if (tid == 0) carry_s = total;
        __syncthreads();
    }
    if (tid == 0) start[n] = carry_s;
}

// ---------------------------------------------------------------------------
// Phase 3: scatter edge source ids into destination-sorted order
// ---------------------------------------------------------------------------
__global__ void gnn_scatter(const int* __restrict__ edges,
                            int* __restrict__ cursor,
                            int* __restrict__ sorted, int nE) {
    int e = blockIdx.x * blockDim.x + threadIdx.x;
    if (e < nE) {
        const int2 de = ((const int2*)edges)[e];          // (dst, src)
        int pos = atomicAdd(&cursor[de.x], 1);
        sorted[pos] = de.y;
    }
}

// ---------------------------------------------------------------------------
// Phase 4: one wave32 per node. Lane l owns float4 #l of the 128-float row.
// Index chunks are double-buffered into per-wave LDS slots with CDNA5
// GLOBAL_LOAD_ASYNC_TO_LDS_B32 (ASYNCcnt), hiding index latency behind the
// 32 row gathers of the current chunk. Rows broadcast via wave32 __shfl;
// each edge costs one coalesced global_load_b128 per lane + 4 VALU adds.
// ---------------------------------------------------------------------------
__global__ __launch_bounds__(128)
void gnn_gather_sum(const float* __restrict__ x,
                    const int* __restrict__ start,
                    const int* __restrict__ sorted,
                    float* __restrict__ out, int nNodes) {
    __shared__ int stage[4][2][32];    // [wave][buffer][lane], 1 KB

    const int lane = threadIdx.x & 31;
    const int wave = threadIdx.x >> 5;
    const int node = blockIdx.x * (blockDim.x >> 5) + wave;
    if (node >= nNodes) return;

    const int s0  = start[node];
    const int s1  = start[node + 1];
    const int deg = s1 - s0;

    const float4* __restrict__ x4 = (const float4*)x;
    float4 acc = make_float4(0.f, 0.f, 0.f, 0.f);

    if (deg > 0) {
        const int nChunks = (deg + 31) >> 5;
        // LDS byte offsets of this lane's slot in each buffer
        // (generic-pointer low 32 bits == LDS offset on gfx1250)
        const unsigned ldsA = (unsigned)(size_t)(&stage[wave][0][lane]);
        const unsigned ldsB = (unsigned)(size_t)(&stage[wave][1][lane]);

        // async-stage chunk 0 into buffer 0
        {
            unsigned voff = (unsigned)(s0 + lane) * 4u;
            asm volatile("global_load_async_to_lds_b32 %0, %1, %2"
                         :: "v"(ldsA), "v"(voff), "s"(sorted) : "memory");
        }

        for (int c = 0; c < nChunks; ++c) {
            const int base = s0 + (c << 5);
            const int cnt  = min(32, s1 - base);
            if (c + 1 < nChunks) {
                // kick off next chunk's indices, then wait for current (oldest)
                const unsigned ldsNext = ((c + 1) & 1) ? ldsB : ldsA;
                unsigned voff = (unsigned)(base + 32 + lane) * 4u;
                asm volatile("global_load_async_to_lds_b32 %0, %1, %2"
                             :: "v"(ldsNext), "v"(voff), "s"(sorted) : "memory");
                asm volatile("s_wait_asynccnt 0x1" ::: "memory");
            } else {
                asm volatile("s_wait_asynccnt 0x0" ::: "memory");
            }

            const int myidx = stage[wave][c & 1][lane];   // ds_load_b32
            if (lane < cnt)                               // warm my row segment
                __builtin_prefetch(&x4[(size_t)myidx * GNN_F4 + lane], 0, 1);

            for (int j = 0; j < cnt; ++j) {
                const int s = __shfl(myidx, j, 32);              // wave32 bcast
                const float4 v = x4[(size_t)s * GNN_F4 + lane];  // b128 gather
                acc.x += v.x; acc.y += v.y; acc.z += v.z; acc.w += v.w;
            }
        }
    }
    ((float4*)out)[(size_t)node * GNN_F4 + lane] = acc;   // global_store_b128
}

// ---------------------------------------------------------------------------
// Fallback (workspace too small): wave-per-edge atomic scatter-add
// ---------------------------------------------------------------------------
__global__ __launch_bounds__(256)
void gnn_atomic_scatter(const float* __restrict__ x,
                        const int* __restrict__ edges,
                        float* __restrict__ out, int nE) {
    const int lane = threadIdx.x & 31;
    const int wave = threadIdx.x >> 5;
    const int e = blockIdx.x * (blockDim.x >> 5) + wave;
    if (e >= nE) return;

    const int2 de = ((const int2*)edges)[e];              // (dst, src)
    const float4 v = ((const float4*)x)[(size_t)de.y * GNN_F4 + lane];
    float* o = out + (size_t)de.x * GNN_FEAT + lane * 4;
    atomicAdd(o + 0, v.x);
    atomicAdd(o + 1, v.y);
    atomicAdd(o + 2, v.z);
    atomicAdd(o + 3, v.w);
}

// ---------------------------------------------------------------------------
extern "C" void kernel_launch(void* const* d_in, const int* in_sizes, int n_in,
                              void* d_out, int out_size, void* d_ws, size_t ws_size,
                              hipStream_t stream) {
    const float* x     = (const float*)d_in[0];
    const int*   edges = (const int*)d_in[1];   // [nE][2] int32: (dst, src)
    float*       out   = (float*)d_out;

    const int nN = in_sizes[0] / GNN_FEAT;      // 10000
    const int nE = in_sizes[1] / 2;             // 640000

    // workspace layout (bytes):
    //   [0,            40004)  start  : (nN+1) ints  (counts -> excl. offsets)
    //   [40960,        80960)  cursor : nN ints
    //   [81920, 81920+4*nE )   sorted : nE ints (src ids sorted by dst)
    //   +128 B padding so tail-chunk async index reads stay in-bounds
    const size_t off_cursor = 40960;
    const size_t off_sorted = 81920;
    const size_t need = off_sorted + (size_t)nE * sizeof(int) + 128;

    if (ws_size >= need) {
        int* start  = (int*)((char*)d_ws);
        int* cursor = (int*)((char*)d_ws + off_cursor);
        int* sorted = (int*)((char*)d_ws + off_sorted);

        gnn_zero_i32<<<(nN + 1 + 255) / 256, 256, 0, stream>>>(start, nN + 1);
        gnn_hist<<<(nE + 255) / 256, 256, 0, stream>>>(edges, start, nE);
        gnn_scan<<<1, 1024, 0, stream>>>(start, cursor, nN);
        gnn_scatter<<<(nE + 255) / 256, 256, 0, stream>>>(edges, cursor, sorted, nE);

        const int wavesPerBlk = 4;                       // 128 threads
        const int blocks = (nN + wavesPerBlk - 1) / wavesPerBlk;
        gnn_gather_sum<<<blocks, wavesPerBlk * 32, 0, stream>>>(x, start, sorted, out, nN);
    } else {
        // fallback: zero output, then float-atomic scatter-add
        gnn_zero_f32<<<((size_t)nN * GNN_FEAT + 255) / 256, 256, 0, stream>>>(out, nN * GNN_FEAT);
        const int blocks = (nE + 7) / 8;                 // 8 waves (edges) per 256-thread block
        gnn_atomic_scatter<<<blocks, 256, 0, stream>>>(x, edges, out, nE);
    }
}